// PIT_MSE_complex_89343909691684
// MI455X (gfx1250) — compile-verified
//
#include <hip/hip_runtime.h>
#include <stdint.h>

// Problem constants (from setup_inputs)
#define S0 2
#define B0 4
#define M0 4
#define W0 3
#define F0 257
#define T0 512
#define C0 4

typedef float v2f __attribute__((ext_vector_type(2)));
typedef float v8f __attribute__((ext_vector_type(8)));

__device__ __forceinline__ v2f ntload2(const v2f* p) {
    return __builtin_nontemporal_load(p);
}

// CDNA5 async global->LDS copy: each lane copies 16B from its global address
// to its workgroup-relative LDS byte offset. Tracked by ASYNCcnt.
__device__ __forceinline__ void async_copy_b128(uint32_t lds_byte_off, const float* gsrc) {
    asm volatile("global_load_async_to_lds_b128 %0, %1, off"
                 :: "v"(lds_byte_off), "v"((uint64_t)(uintptr_t)gsrc)
                 : "memory");
}
__device__ __forceinline__ void wait_async0() {
#if __has_builtin(__builtin_amdgcn_s_wait_asynccnt)
    __builtin_amdgcn_s_wait_asynccnt(0);
#else
    asm volatile("s_wait_asynccnt 0" ::: "memory");
#endif
}

// Kernel 1: one block per (s,b,f). Stages mixture rows in LDS via async copy,
// computes filtered output + per-target L1/mag losses, block-reduces, writes
// partial[((s*B+b)*F+f)*2 + s_tgt].
__global__ __launch_bounds__(256)
void pit_filt_loss_kernel(const float* __restrict__ masks,
                          const float* __restrict__ mixr,
                          const float* __restrict__ mixi,
                          const float* __restrict__ tgtr,
                          const float* __restrict__ tgti,
                          float* __restrict__ partial) {
    __shared__ float mixlds[2 * M0 * T0];   // [comp][m][t], 16 KB
    __shared__ float red[256];

    const int tid = threadIdx.x;
    const int blk = blockIdx.x;             // (s*B0 + b)*F0 + f
    const int f   = blk % F0;
    const int sb  = blk / F0;               // s*B0 + b
    const int b   = sb % B0;
    const int s   = sb / B0;

    // ---- stage 8 rows (4 mics x r/i x 512 f32) into LDS: 1024 x 16B chunks ----
    #pragma unroll
    for (int k = 0; k < 4; ++k) {
        int c    = k * 256 + tid;           // 0..1023
        int row  = c >> 7;                  // 0..7  (comp*4 + m)
        int pos  = (c & 127) << 2;          // float offset in row, 16B-aligned
        int comp = row >> 2;
        int m    = row & 3;
        const float* src = (comp ? mixi : mixr)
                         + (((size_t)(sb) * M0 + m) * F0 + f) * (size_t)T0 + pos;
        uint32_t loff = (uint32_t)(uintptr_t)(&mixlds[row * T0 + pos]);
        async_copy_b128(loff, src);
    }
    wait_async0();
    __syncthreads();

    const v2f* mp = (const v2f*)masks;      // complex pairs
    float s0 = 0.f, s1 = 0.f;

    #pragma unroll
    for (int it = 0; it < 2; ++it) {
        const int t = tid + it * 256;
        float orr = 0.f, oii = 0.f;
        #pragma unroll
        for (int m = 0; m < M0; ++m) {
            const float* lr = &mixlds[m * T0];
            const float* li = &mixlds[(M0 + m) * T0];
            const int mb = ((sb * M0 + m) * W0) * F0;   // mask pair-index prefix
            #pragma unroll
            for (int w = 0; w < W0; ++w) {
                const int tt = t + w - 1;               // p = (W-1)/2 = 1
                float cr = 0.f, ci = 0.f;
                if (tt >= 0 && tt < T0) { cr = lr[tt]; ci = li[tt]; }
                v2f fm = ntload2(&mp[(size_t)(mb + w * F0 + f) * T0 + t]);
                // out += mix * filt   (complex)
                orr = fmaf(cr, fm.x, fmaf(-ci, fm.y, orr));
                oii = fmaf(cr, fm.y, fmaf( ci, fm.x, oii));
            }
        }
        const float om = sqrtf(orr * orr + oii * oii);
        #pragma unroll
        for (int st = 0; st < S0; ++st) {
            const size_t tix = ((size_t)(st * B0 + b) * C0 * F0 + f) * (size_t)T0 + t;
            const float tr = tgtr[tix], ti = tgti[tix];
            const float l  = fabsf(tr - orr) + fabsf(ti - oii)
                           + fabsf(sqrtf(tr * tr + ti * ti) - om);
            if (st == 0) s0 += l; else s1 += l;
        }
    }

    // ---- deterministic block tree reduction for both target sums ----
    #pragma unroll
    for (int j = 0; j < 2; ++j) {
        red[tid] = (j == 0) ? s0 : s1;
        __syncthreads();
        for (int off = 128; off > 0; off >>= 1) {
            if (tid < off) red[tid] += red[tid + off];
            __syncthreads();
        }
        if (tid == 0) partial[(size_t)blk * 2 + j] = red[0];
        __syncthreads();
    }
}

// Kernel 2 (single wave32): WMMA row-sum reduction over f for the 16
// (s_out,b,s_tgt) combos, then permutation-min + batch sum + /num_utts.
__global__ __launch_bounds__(32)
void pit_final_kernel(const float* __restrict__ partial,
                      float* __restrict__ out,
                      float inv_num_utts) {
    __shared__ float Lsh[16];

    const int lane  = threadIdx.x;          // 0..31
    const int combo = lane & 15;            // M index: s_out*8 + b*2 + s_tgt
    const int kbase = (lane < 16) ? 0 : 2;  // A layout: lanes16-31 hold K=2,3

    v8f acc = {};
    v2f bones; bones.x = 1.f; bones.y = 1.f;  // B = all ones -> C[:,n] = row sums

    for (int it = 0; it < 65; ++it) {       // covers f = 0..259 (guard F0=257)
        const int fA = it * 4 + kbase;
        v2f a;
        a.x = (fA     < F0) ? partial[((size_t)(combo >> 1) * F0 + fA    ) * 2 + (combo & 1)] : 0.f;
        a.y = (fA + 1 < F0) ? partial[((size_t)(combo >> 1) * F0 + fA + 1) * 2 + (combo & 1)] : 0.f;
        acc = __builtin_amdgcn_wmma_f32_16x16x4_f32(
                  false, a, false, bones, (short)0, acc, false, false);
    }

    // Column N=0 of C: lane 0 holds M=0..7 (VGPR r), lane 16 holds M=8..15.
    if (lane == 0) {
        #pragma unroll
        for (int r = 0; r < 8; ++r) Lsh[r] = acc[r];
    }
    if (lane == 16) {
        #pragma unroll
        for (int r = 0; r < 8; ++r) Lsh[8 + r] = acc[r];
    }
    __syncthreads();

    if (lane == 0) {
        float tot = 0.f;
        #pragma unroll
        for (int b = 0; b < B0; ++b) {
            // combo = s_out*8 + b*2 + s_tgt ; pscore = 3*sum/S = 1.5*(..)
            const float pid = 1.5f * (Lsh[0 * 8 + b * 2 + 0] + Lsh[1 * 8 + b * 2 + 1]);
            const float psw = 1.5f * (Lsh[0 * 8 + b * 2 + 1] + Lsh[1 * 8 + b * 2 + 0]);
            tot += fminf(pid, psw);
        }
        out[0] = tot * inv_num_utts;
    }
}

extern "C" void kernel_launch(void* const* d_in, const int* in_sizes, int n_in,
                              void* d_out, int out_size, void* d_ws, size_t ws_size,
                              hipStream_t stream) {
    const float* masks = (const float*)d_in[0];
    const float* mixr  = (const float*)d_in[1];
    const float* mixi  = (const float*)d_in[2];
    const float* tgtr  = (const float*)d_in[3];
    const float* tgti  = (const float*)d_in[4];
    const float  inv_num_utts = 1.0f / (float)in_sizes[5];   // num_utts = B

    float* partial = (float*)d_ws;   // S*B*F*2 floats = 16448 B, fully overwritten

    pit_filt_loss_kernel<<<dim3(S0 * B0 * F0), dim3(256), 0, stream>>>(
        masks, mixr, mixi, tgtr, tgti, partial);
    pit_final_kernel<<<dim3(1), dim3(32), 0, stream>>>(
        partial, (float*)d_out, inv_num_utts);
}